// ANNOutputTorch_57913339019800
// MI455X (gfx1250) — compile-verified
//
#include <hip/hip_runtime.h>

// CDNA5 wave32 WMMA types
typedef __attribute__((ext_vector_type(2))) float v2f;
typedef __attribute__((ext_vector_type(8))) float v8f;

#define N_ATOMS       8388608
#define OUT_UNITS     16
#define N_STRUCT      4096
#define PAD_SEG       (N_STRUCT + 16)           // rebase window may run 15 past the end
#define ROWS_PER_WAVE 1024
#define NWAVES        (N_ATOMS / ROWS_PER_WAVE) // 8192
#define BLOCK_THREADS 256
#define WAVES_PER_BLOCK (BLOCK_THREADS / 32)

__global__ void zero_sums_kernel(float* __restrict__ sums) {
    int i = blockIdx.x * blockDim.x + threadIdx.x;
    if (i < PAD_SEG * OUT_UNITS) sums[i] = 0.0f;
}

// Each wave streams a contiguous chunk of rows. A 16-row x 16-col f32 WMMA
// accumulator holds partial sums for a 16-wide window of segment values
// [seg_base, seg_base+16). A-matrix is a one-hot selector: A[i][k] = (i ==
// ind[row_k] - seg_base), so D += A*B bins 4 data rows per WMMA into their
// segments. Window rebases (sorted indices -> rare) flush via f32 atomics.
__global__ void __launch_bounds__(BLOCK_THREADS)
seg_sum_wmma_kernel(const int* __restrict__ ind,
                    const float* __restrict__ data,
                    float* __restrict__ sums) {
    const int lane = threadIdx.x & 31;
    const int wave = (blockIdx.x * BLOCK_THREADS + threadIdx.x) >> 5;
    const int half = lane >> 4;    // which 16-lane half
    const int m    = lane & 15;    // row index of A / column index of B,D

    const int r0 = wave * ROWS_PER_WAVE;

    v8f acc = {0.f, 0.f, 0.f, 0.f, 0.f, 0.f, 0.f, 0.f};
    int seg_base = __builtin_amdgcn_readlane(ind[r0], 0);  // uniform (SGPR)

    // 32-bit strength-reduced offsets (max byte offset 512MB < 4GB):
    // saddr + voffset addressing, no 64-bit VALU address math in the loop.
    unsigned ioff = (unsigned)(r0 + 2 * half);            // element offset into ind
    unsigned doff = (unsigned)(r0 + 2 * half) * OUT_UNITS + m; // element offset into data
    int r = r0;
    const int rend = r0 + ROWS_PER_WAVE;

#pragma unroll 2
    for (; r < rend; r += 8, ioff += 8, doff += 8 * OUT_UNITS) {
        // indices for this lane's two K slots in each 4-row micro-tile
        // group0 rows: r+2*half, r+2*half+1 ; group1 rows: +4
        int2 iA = *reinterpret_cast<const int2*>(ind + ioff);
        int2 iB = *reinterpret_cast<const int2*>(ind + ioff + 4);
        int s0 = __builtin_amdgcn_readlane(iA.x, 0);   // ind[r]   (uniform SGPR)
        int s7 = __builtin_amdgcn_readlane(iB.y, 31);  // ind[r+7] (uniform, sorted -> max)

        if (__builtin_expect(s7 - s0 >= 16, 0)) {
            // pathological value gap inside 8 rows: direct-atomic fallback
            for (int t = lane; t < 8 * OUT_UNITS; t += 32) {
                int rr = t >> 4, cc = t & 15;
                int sg = ind[r + rr];
                atomicAdd(&sums[sg * OUT_UNITS + cc],
                          data[(unsigned)(r + rr) * OUT_UNITS + cc]);
            }
            continue;
        }
        if (__builtin_expect(s7 - seg_base >= 16, 0)) {
            // window exhausted: flush nonzero accumulator rows, rebase
#pragma unroll
            for (int j = 0; j < 8; ++j) {
                int row = seg_base + j + 8 * half;  // D layout: VGPR j = rows j, j+8
                if (acc[j] != 0.0f)
                    atomicAdd(&sums[row * OUT_UNITS + m], acc[j]);
                acc[j] = 0.0f;
            }
            seg_base = s0;
        }

        // A: one-hot selectors (A-layout: lanes0-15 K={0,1}, lanes16-31 K={2,3})
        v2f A0, A1, B0, B1;
        A0.x = (m == (iA.x - seg_base)) ? 1.0f : 0.0f;
        A0.y = (m == (iA.y - seg_base)) ? 1.0f : 0.0f;
        A1.x = (m == (iB.x - seg_base)) ? 1.0f : 0.0f;
        A1.y = (m == (iB.y - seg_base)) ? 1.0f : 0.0f;

        // B: data rows (B-layout: VGPR0 = {K0,K2}, VGPR1 = {K1,K3} across halves)
        B0.x = data[doff];
        B0.y = data[doff + OUT_UNITS];
        B1.x = data[doff + 4 * OUT_UNITS];
        B1.y = data[doff + 5 * OUT_UNITS];

        acc = __builtin_amdgcn_wmma_f32_16x16x4_f32(false, A0, false, B0,
                                                    (short)0, acc, false, false);
        acc = __builtin_amdgcn_wmma_f32_16x16x4_f32(false, A1, false, B1,
                                                    (short)0, acc, false, false);
    }

    // final flush
#pragma unroll
    for (int j = 0; j < 8; ++j) {
        int row = seg_base + j + 8 * half;
        if (acc[j] != 0.0f)
            atomicAdd(&sums[row * OUT_UNITS + m], acc[j]);
    }
}

// Sorted indices -> per-segment count via two binary searches per segment.
__global__ void seg_count_kernel(const int* __restrict__ ind,
                                 float* __restrict__ cnts) {
    int s = blockIdx.x * blockDim.x + threadIdx.x;
    if (s >= N_STRUCT) return;
    int lo0 = 0, hi0 = N_ATOMS;
    while (lo0 < hi0) { int mid = (lo0 + hi0) >> 1; if (ind[mid] < s)     lo0 = mid + 1; else hi0 = mid; }
    int lo1 = lo0, hi1 = N_ATOMS;
    while (lo1 < hi1) { int mid = (lo1 + hi1) >> 1; if (ind[mid] < s + 1) lo1 = mid + 1; else hi1 = mid; }
    cnts[s] = (float)(lo1 - lo0);
}

__global__ void finalize_kernel(const float* __restrict__ sums,
                                const float* __restrict__ cnts,
                                float* __restrict__ out) {
    int i = blockIdx.x * blockDim.x + threadIdx.x;
    if (i >= N_STRUCT * OUT_UNITS) return;
    out[i] = sums[i] / fmaxf(cnts[i >> 4], 1.0f);
}

extern "C" void kernel_launch(void* const* d_in, const int* in_sizes, int n_in,
                              void* d_out, int out_size, void* d_ws, size_t ws_size,
                              hipStream_t stream) {
    const int*   ind  = (const int*)d_in[0];    // ind_1 (sorted segment ids)
    const float* data = (const float*)d_in[1];  // output (N_ATOMS x 16 f32)
    float* sums = (float*)d_ws;                       // PAD_SEG*16 floats
    float* cnts = sums + (size_t)PAD_SEG * OUT_UNITS; // N_STRUCT floats
    float* out  = (float*)d_out;

    zero_sums_kernel<<<(PAD_SEG * OUT_UNITS + 255) / 256, 256, 0, stream>>>(sums);
    seg_sum_wmma_kernel<<<NWAVES / WAVES_PER_BLOCK, BLOCK_THREADS, 0, stream>>>(ind, data, sums);
    seg_count_kernel<<<(N_STRUCT + 255) / 256, 256, 0, stream>>>(ind, cnts);
    finalize_kernel<<<(N_STRUCT * OUT_UNITS + 255) / 256, 256, 0, stream>>>(sums, cnts, out);
}